// ManualConv2d_4020089389279
// MI455X (gfx1250) — compile-verified
//
#include <hip/hip_runtime.h>

typedef __attribute__((ext_vector_type(2))) float v2f;
typedef __attribute__((ext_vector_type(4))) float v4f;
typedef __attribute__((ext_vector_type(8))) float v8f;

#define HH 512
#define WW 512
#define CC 16          // in & out channels
#define PXS 20         // padded LDS pixel stride (floats): conflict-free & 16B aligned
#define PATCHW 130     // 128-pixel block tile + 1 halo each side
#define FCO 18         // LDS filter ci-stride (floats): 8B aligned, conflict-free-ish

__global__ __launch_bounds__(256)
void conv3x3_wmma_f32(const float* __restrict__ x,
                      const float* __restrict__ f,
                      const float* __restrict__ bias,
                      float* __restrict__ out)
{
    __shared__ float sIn[3 * PATCHW * PXS];   // 3*130*20*4 = 31,200 B
    __shared__ float sF[9 * 16 * FCO];        // 9*16*18*4  = 10,368 B

    const int tid  = threadIdx.x;
    const int b    = blockIdx.x;
    const int wseg = b & 3;              // 4 x 128-pixel segments per row
    const int h    = (b >> 2) & 511;
    const int n    = b >> 11;
    const int w0   = wseg * 128;

    // ---- stage filters transposed: sF[(tap*16+co)*FCO + ci] = f[(tap*16+ci)*16 + co]
    for (int i = tid; i < 9 * 16 * 16; i += 256) {
        int co  = i & 15;
        int ci  = (i >> 4) & 15;
        int tap = i >> 8;
        sF[(tap * 16 + co) * FCO + ci] = f[(tap * 16 + ci) * 16 + co];
    }

    // ---- stage input patch: rows h-1..h+1, pixels w0-1..w0+128, 16 ch (4 x float4/px)
    for (int i = tid; i < 3 * PATCHW * 4; i += 256) {
        int c4 = i & 3;
        int px = (i >> 2) % PATCHW;
        int r  = (i >> 2) / PATCHW;
        int gh = h + r - 1;
        int gw = w0 + px - 1;
        v4f val = {0.f, 0.f, 0.f, 0.f};
        if ((unsigned)gh < (unsigned)HH && (unsigned)gw < (unsigned)WW) {
            val = *(const v4f*)&x[(((n * HH + gh) * WW + gw) * CC) + c4 * 4];
        }
        *(v4f*)&sIn[(r * PATCHW + px) * PXS + c4 * 4] = val;
    }
    __syncthreads();

    const int lane = tid & 31;
    const int wid  = tid >> 5;          // wave id: which 16-pixel tile in the block
    const int half = lane >> 4;         // 0: K=0..1 side, 1: K=2..3 side
    const int m    = lane & 15;         // A: M index / B: N index

    v8f acc = {};                       // 16x16 f32 accumulator (8 VGPRs)

#pragma unroll
    for (int dy = 0; dy < 3; ++dy) {
#pragma unroll
        for (int dx = 0; dx < 3; ++dx) {
            const int tap = dy * 3 + dx;
            // A: lane holds pixel (wid*16 + m + dx) of LDS row dy, channels kc+2*half+{0,1}
            const float* aBase = &sIn[(dy * PATCHW + (wid * 16 + m + dx)) * PXS + 2 * half];
            // B: lane holds co=m, ci = kc+2*half+{0,1}
            const float* bBase = &sF[(tap * 16 + m) * FCO + 2 * half];
#pragma unroll
            for (int kc = 0; kc < 16; kc += 4) {
                v2f a  = *(const v2f*)(aBase + kc);
                v2f bm = *(const v2f*)(bBase + kc);
                acc = __builtin_amdgcn_wmma_f32_16x16x4_f32(
                          /*neg_a=*/false, a, /*neg_b=*/false, bm,
                          /*c_mod=*/(short)0, acc,
                          /*reuse_a=*/false, /*reuse_b=*/false);
            }
        }
    }

    // ---- bias + store: acc[r] is (M = r + 8*half, N = m)
    const float bv      = bias[m];
    const int   pixBase = (n * HH + h) * WW + (w0 + wid * 16) + 8 * half;
#pragma unroll
    for (int r = 0; r < 8; ++r) {
        out[(pixBase + r) * CC + m] = acc[r] + bv;
    }
}

extern "C" void kernel_launch(void* const* d_in, const int* in_sizes, int n_in,
                              void* d_out, int out_size, void* d_ws, size_t ws_size,
                              hipStream_t stream) {
    (void)in_sizes; (void)n_in; (void)out_size; (void)d_ws; (void)ws_size;
    const float* x    = (const float*)d_in[0];   // (16,512,512,16) f32
    const float* filt = (const float*)d_in[1];   // (3,3,16,16) f32
    const float* bias = (const float*)d_in[2];   // (16,) f32
    float*       out  = (float*)d_out;           // (16,512,512,16) f32

    dim3 grid(16 * 512 * 4);   // n * h * (512/128)
    dim3 block(256);           // 8 waves, each one 16x16 output tile
    conv3x3_wmma_f32<<<grid, block, 0, stream>>>(x, filt, bias, out);
}